// SparseGraphConvolution_69389491634674
// MI455X (gfx1250) — compile-verified
//
#include <hip/hip_runtime.h>

typedef float v2f __attribute__((ext_vector_type(2)));
typedef float v8f __attribute__((ext_vector_type(8)));
typedef unsigned int u32x4 __attribute__((ext_vector_type(4)));
typedef int i32x4 __attribute__((ext_vector_type(4)));
typedef int i32x8 __attribute__((ext_vector_type(8)));

#define H_ 4
#define T_ 8
#define N_ 1024
#define M_ 256
#define C_ 16
#define E_ 16

// ---------------------------------------------------------------------------
// Kernel 1: edge[t,h,m,c] = sum_e edge_init[t,h,m,e] * Wge[c,e] + bge[c]
//           div[t,n]      = sum_m G[t,m,n]
// ---------------------------------------------------------------------------
__global__ __launch_bounds__(256) void prep_edge_div(
    const float* __restrict__ edge_init, const float* __restrict__ Wge,
    const float* __restrict__ bge, const float* __restrict__ G,
    float* __restrict__ edge, float* __restrict__ divArr)
{
  int gid = blockIdx.x * 256 + threadIdx.x;
  if (blockIdx.x < 512) {               // 512*256 = 131072 = T*H*M*C
    int row = gid >> 4, c = gid & 15;
    float s = bge[c];
#pragma unroll
    for (int e = 0; e < 16; ++e) s += edge_init[row * 16 + e] * Wge[c * 16 + e];
    edge[gid] = s;
  } else {                              // 32*256 = 8192 = T*N
    int idx = gid - 512 * 256;
    int t = idx >> 10, n = idx & 1023;
    float s = 0.f;
    for (int m = 0; m < M_; ++m) s += G[((size_t)t * M_ + m) * N_ + n];
    divArr[idx] = s;
  }
}

// ---------------------------------------------------------------------------
// Kernel 2: spa[t,h,n,c] = graph[h,t,n,1+c]
//           val[t,h,n,c] = spa + (sum_m G[t,m,n]*edge[t,h,m,c]) / div[t,n]
// ---------------------------------------------------------------------------
__global__ __launch_bounds__(256) void prep_value(
    const float* __restrict__ graph, const float* __restrict__ G,
    const float* __restrict__ edge, const float* __restrict__ divArr,
    float* __restrict__ spa, float* __restrict__ val)
{
  int bid = blockIdx.x;                 // 2048 = 32 th * 64 n-blocks
  int th = bid >> 6, nb = bid & 63;
  int t = th >> 2, h = th & 3;
  int c = threadIdx.x & 15, nl = threadIdx.x >> 4;
  int n = nb * 16 + nl;
  float sv = graph[((size_t)(h * T_ + t) * N_ + n) * (C_ + 1) + 1 + c];
  float s = 0.f;
  for (int m = 0; m < M_; ++m)
    s += G[((size_t)t * M_ + m) * N_ + n] * edge[((size_t)th * M_ + m) * C_ + c];
  size_t o = (size_t)th * N_ * C_ + (size_t)n * C_ + c;
  spa[o] = sv;
  val[o] = sv + s / divArr[t * N_ + n];
}

// ---------------------------------------------------------------------------
// TDM: DMA one contiguous [1024x16] f32 panel (64 KB) from global into LDS.
// D# group0: count=1, lds_addr, global_addr, type=2 (image)
// D# group1: data_size=4B, tensor_dim0=16384, tensor_dim1=1, tile_dim0=16384,
//            tile_dim1=1, tensor_dim0_stride=16384. Groups 2/3 zero (<=2D).
// Issued by one wave; completion via s_wait_tensorcnt 0, then barrier.
// ---------------------------------------------------------------------------
__device__ __forceinline__ void tdm_load_panel(const float* gsrc,
                                               unsigned lds_byte_off) {
#if __has_builtin(__builtin_amdgcn_tensor_load_to_lds)
  unsigned long long ga = (unsigned long long)(uintptr_t)gsrc;
  u32x4 g0;
  g0.x = 1u;                                                  // count = 1
  g0.y = lds_byte_off;                                        // LDS dest (bytes)
  g0.z = (unsigned)(ga & 0xffffffffu);                        // global_addr lo
  g0.w = (unsigned)((ga >> 32) & 0x01ffffffu) | 0x80000000u;  // addr hi | type=2
  i32x8 g1;
  g1.s0 = (int)(2u << 16);          // workgroup_mask=0, data_size=2 (4 bytes)
  g1.s1 = (int)(16384u << 16);      // tensor_dim0[15:0]=16384 in bits[63:48]
  g1.s2 = (int)(1u << 16);          // tensor_dim0 hi=0; tensor_dim1=1
  g1.s3 = (int)(16384u << 16);      // tile_dim0 = 16384 in bits[127:112]
  g1.s4 = 1;                        // tile_dim1 = 1, tile_dim2 = 0
  g1.s5 = 16384;                    // tensor_dim0_stride = 16384 elements
  g1.s6 = 0;
  g1.s7 = 0;
  i32x4 gz4 = {0, 0, 0, 0};
  i32x8 gz8 = {0, 0, 0, 0, 0, 0, 0, 0};
  __builtin_amdgcn_tensor_load_to_lds(g0, g1, gz4, gz4, gz8, 0);
  __builtin_amdgcn_s_wait_tensorcnt(0);
#endif
}

// ---------------------------------------------------------------------------
// Kernel 3: per (t,h): pair = P @ spa ; grp = Gi @ val  (both [1024x16], K=1024)
// then out[n,h,t,e] = prelu(pair@Wp^T) + prelu(grp@Wg^T)
// 256 blocks = 32 (t,h) x 8 row-blocks of 128; 8 waves/block, 16 rows/wave.
// WMMA f32 16x16x4: A = 16x4 of the streamed NxN matrix, B = 4x16 panel in LDS
// (panel DMA'd in by the Tensor Data Mover, row-major [n*16+c]).
// ---------------------------------------------------------------------------
__global__ __launch_bounds__(256) void fused_gemm(
    const float* __restrict__ P, const float* __restrict__ Gi,
    const float* __restrict__ spa, const float* __restrict__ val,
    const float* __restrict__ Wp, const float* __restrict__ Wg,
    const float* __restrict__ pAp, const float* __restrict__ pAg,
    float* __restrict__ out)
{
  __shared__ float lds[16384];          // 64 KB: B panel, later accum staging
  const int tid = threadIdx.x;
  const int bid = blockIdx.x;
  const int th = bid >> 3;              // 0..31  (t*H + h)
  const int nb = bid & 7;               // 0..7   (row block of 128)
  const int t = th >> 2, h = th & 3;
  const int lane = tid & 31, wave = tid >> 5;
  const int c16 = lane & 15;
  const int kOff = (lane >> 4) * 2;     // lanes 16..31 hold K+2,K+3 of A
  const int row = nb * 128 + wave * 16 + c16;   // A row (M index) for this lane
  const size_t matOff = (size_t)th * N_ * N_;

  // ---- panel 1: spa (TDM async DMA -> LDS) ----
#if __has_builtin(__builtin_amdgcn_tensor_load_to_lds)
  if (wave == 0) tdm_load_panel(spa + (size_t)th * N_ * C_, 0u);
#else
  {
    const float* src = spa + (size_t)th * N_ * C_;
    for (int i = tid; i < N_ * C_; i += 256) lds[i] = src[i];
  }
#endif
  __syncthreads();

  v8f accP = {};
  {
    const float* aptr = P + matOff + (size_t)row * N_ + kOff;
#pragma unroll 8
    for (int k0 = 0; k0 < N_; k0 += 4) {
      if ((k0 & 255) == 0) __builtin_prefetch(aptr + 256, 0, 0);
      v2f a = *(const v2f*)aptr; aptr += 4;
      int kb = k0 + kOff;
      v2f b;
      b.x = lds[kb * 16 + c16];
      b.y = lds[(kb + 1) * 16 + c16];
      accP = __builtin_amdgcn_wmma_f32_16x16x4_f32(
          false, a, false, b, (short)0, accP, false, false);
    }
  }
  __syncthreads();

  // ---- panel 2: value (TDM async DMA -> LDS) ----
#if __has_builtin(__builtin_amdgcn_tensor_load_to_lds)
  if (wave == 0) tdm_load_panel(val + (size_t)th * N_ * C_, 0u);
#else
  {
    const float* src = val + (size_t)th * N_ * C_;
    for (int i = tid; i < N_ * C_; i += 256) lds[i] = src[i];
  }
#endif
  __syncthreads();

  v8f accG = {};
  {
    const float* aptr = Gi + matOff + (size_t)row * N_ + kOff;
#pragma unroll 8
    for (int k0 = 0; k0 < N_; k0 += 4) {
      if ((k0 & 255) == 0) __builtin_prefetch(aptr + 256, 0, 0);
      v2f a = *(const v2f*)aptr; aptr += 4;
      int kb = k0 + kOff;
      v2f b;
      b.x = lds[kb * 16 + c16];
      b.y = lds[(kb + 1) * 16 + c16];
      accG = __builtin_amdgcn_wmma_f32_16x16x4_f32(
          false, a, false, b, (short)0, accG, false, false);
    }
  }
  __syncthreads();

  // ---- stage accumulator tiles to LDS (C layout: vgpr r -> M=r / r+8) ----
  {
    int mBase = wave * 16 + ((lane >> 4) * 8);
#pragma unroll
    for (int r = 0; r < 8; ++r) {
      lds[(mBase + r) * 16 + c16] = accP[r];
      lds[2048 + (mBase + r) * 16 + c16] = accG[r];
    }
  }
  __syncthreads();

  // ---- epilogue: E-projection + PReLU + transpose to [N,H,T,E] ----
  const float ap = *pAp, ag = *pAg;
  const int e = tid & 15;
  const int ml0 = tid >> 4;
  float wp[16], wg[16];
#pragma unroll
  for (int c = 0; c < 16; ++c) {
    wp[c] = Wp[e * 16 + c];
    wg[c] = Wg[e * 16 + c];
  }
#pragma unroll
  for (int j = 0; j < 8; ++j) {
    int mLocal = ml0 + j * 16;
    float sp = 0.f, sg = 0.f;
#pragma unroll
    for (int c = 0; c < 16; ++c) {
      sp += lds[mLocal * 16 + c] * wp[c];
      sg += lds[2048 + mLocal * 16 + c] * wg[c];
    }
    sp = sp >= 0.f ? sp : ap * sp;
    sg = sg >= 0.f ? sg : ag * sg;
    int m = nb * 128 + mLocal;
    out[(((size_t)m * H_ + h) * T_ + t) * E_ + e] = sp + sg;
  }
}

// ---------------------------------------------------------------------------
extern "C" void kernel_launch(void* const* d_in, const int* in_sizes, int n_in,
                              void* d_out, int out_size, void* d_ws, size_t ws_size,
                              hipStream_t stream) {
  const float* graph = (const float*)d_in[0];
  const float* pairI = (const float*)d_in[1];
  const float* grpI  = (const float*)d_in[2];
  const float* G     = (const float*)d_in[3];
  const float* edgeI = (const float*)d_in[4];
  const float* Wp    = (const float*)d_in[5];
  const float* pAp   = (const float*)d_in[6];
  const float* Wge   = (const float*)d_in[7];
  const float* bge   = (const float*)d_in[8];
  const float* Wg    = (const float*)d_in[9];
  const float* pAg   = (const float*)d_in[10];
  float* out = (float*)d_out;
  float* ws  = (float*)d_ws;

  float* edge = ws;                 // T*H*M*C      = 131072 floats
  float* divA = ws + 131072;        // T*N          =   8192
  float* spa  = ws + 139264;        // T*H*N*C      = 524288
  float* val  = ws + 663552;        // T*H*N*C      = 524288

  prep_edge_div<<<544, 256, 0, stream>>>(edgeI, Wge, bge, G, edge, divA);
  prep_value<<<2048, 256, 0, stream>>>(graph, G, edge, divA, spa, val);
  fused_gemm<<<256, 256, 0, stream>>>(pairI, grpI, spa, val, Wp, Wg, pAp, pAg, out);
}